// VQExpertEMA_54090818126065
// MI455X (gfx1250) — compile-verified
//
#include <hip/hip_runtime.h>

#define T_TOK 8192
#define CDIM  512
#define KCODES 8192
#define DECAYF 0.99f
#define ONE_MINUS_DECAY 0.01f
#define BETAF 0.25f
#define EPSF 1e-5f

typedef float v2f __attribute__((ext_vector_type(2)));
typedef float v8f __attribute__((ext_vector_type(8)));

#if __has_builtin(__builtin_amdgcn_tensor_load_to_lds) && __has_builtin(__builtin_amdgcn_s_wait_tensorcnt)
#define HAVE_TDM 1
#else
#define HAVE_TDM 0
#endif

#if HAVE_TDM
typedef unsigned int v4u __attribute__((ext_vector_type(4)));
typedef int v4i __attribute__((ext_vector_type(4)));
typedef int v8i __attribute__((ext_vector_type(8)));

// One 2D TDM descriptor: copy a 16-row x 512-col fp32 tile (row stride 512)
// from global memory to LDS. Issued once per wave (EXEC ignored by TDM).
__device__ __forceinline__ void tdm_load_tile(const float* gsrc, unsigned lds_off) {
    unsigned long long ga = (unsigned long long)(size_t)gsrc;
    v4u g0;
    g0.x = 1u;                                   // [1:0] count=1 (valid), user mode
    g0.y = lds_off;                              // [63:32] lds_addr (bytes)
    g0.z = (unsigned)(ga & 0xffffffffu);         // [95:64] global_addr lo
    g0.w = (unsigned)(ga >> 32) | (2u << 30);    // [120:96] addr hi | [127:126] type=2
    v8i g1;
    g1[0] = (int)(2u << 16);                     // [17:16] data_size = 2 -> 4 bytes
    g1[1] = (int)(512u << 16);                   // [79:48] tensor_dim0 = 512 (lo16)
    g1[2] = (int)(16u << 16);                    // dim0 hi16=0 | [111:80] tensor_dim1=16
    g1[3] = (int)(512u << 16);                   // dim1 hi16=0 | [127:112] tile_dim0=512
    g1[4] = 16;                                  // [143:128] tile_dim1=16 | tile_dim2=0
    g1[5] = 512;                                 // [207:160] tensor_dim0_stride lo32
    g1[6] = 0;                                   // stride hi16 | tensor_dim1_stride lo16
    g1[7] = 0;
    v4i z4 = {0, 0, 0, 0};                       // groups 2/3 unused (2D tensor)
#if __clang_major__ >= 23
    v8i z8 = {0, 0, 0, 0, 0, 0, 0, 0};
    __builtin_amdgcn_tensor_load_to_lds(g0, g1, z4, z4, z8, 0);
#else
    __builtin_amdgcn_tensor_load_to_lds(g0, g1, z4, z4, 0);
#endif
}
#endif

// ---------- ||e_k||^2 : one wave per codebook row ----------
__global__ void k_en2(const float* __restrict__ cb, float* __restrict__ en2) {
    int k = blockIdx.x;
    int lane = threadIdx.x;
    const float* row = cb + (size_t)k * CDIM;
    float s = 0.f;
    for (int c = lane; c < CDIM; c += 32) { float v = row[c]; s += v * v; }
    for (int m = 16; m >= 1; m >>= 1) s += __shfl_xor(s, m, 32);
    if (lane == 0) en2[k] = s;
}

// ---------- init EMA outputs (DECAY * old) + zero loss ----------
__global__ void k_init(const float* __restrict__ embed_avg,
                       const float* __restrict__ cluster_size,
                       float* __restrict__ out_ea, float* __restrict__ out_cs,
                       float* __restrict__ out_loss) {
    int i = blockIdx.x * blockDim.x + threadIdx.x;
    if (i < KCODES * CDIM) out_ea[i] = embed_avg[i] * DECAYF;
    if (i < KCODES)        out_cs[i] = cluster_size[i] * DECAYF;
    if (i == 0)            *out_loss = 0.f;
}

// ---------- fused distance-GEMM + argmin via V_WMMA_F32_16X16X4_F32 ----------
// 64 workgroups x 256 threads (8 waves). Wave w owns tokens [wg*128+w*16, +16).
// dist(t,k) = ||e_k||^2 - 2 * z_t . e_k   (||z||^2 dropped: constant per row)
// Codebook tiles double-buffered and streamed by the Tensor Data Mover.
__launch_bounds__(256, 1)
__global__ void k_argmin(const float* __restrict__ z,
                         const float* __restrict__ cb,
                         const float* __restrict__ en2,
                         int* __restrict__ idx_out,
                         float* __restrict__ idx_out_f) {
    __shared__ __align__(16) float s_z[128 * CDIM];      // 256 KB persistent token slab
    __shared__ __align__(16) float s_cb[2][16 * CDIM];   // 2 x 32 KB streamed tiles

    const int tid = threadIdx.x;
    const int wg  = blockIdx.x;
    const int NT  = KCODES / 16;

    // stage 128-token z slab (contiguous 256 KB) into LDS
    {
        const float4* src = (const float4*)(z + (size_t)wg * 128 * CDIM);
        float4* dst = (float4*)s_z;
        for (int i = tid; i < 128 * CDIM / 4; i += 256) dst[i] = src[i];
    }

#if HAVE_TDM
    // prologue: wave 0 launches TDM DMAs for tiles 0 and 1
    if (tid < 32) {
        tdm_load_tile(cb, (unsigned)(size_t)(void*)&s_cb[0][0]);
        tdm_load_tile(cb + (size_t)16 * CDIM, (unsigned)(size_t)(void*)&s_cb[1][0]);
    }
#endif
    __syncthreads();

    const int wave = tid >> 5;
    const int lane = tid & 31;
    const int hl   = lane >> 4;   // half of the wave
    const int mn   = lane & 15;   // row (A) / column (B,D) index

    // A frag: lane holds token row (wave*16+mn); VGPR pair = K {hl*2, hl*2+1}
    const float* aBase = s_z + ((size_t)((wave << 4) + mn)) * CDIM + hl * 2;
    const int bOff = mn * CDIM + hl * 2;

    float minv[8];
    int   mini[8];
#pragma unroll
    for (int i = 0; i < 8; ++i) { minv[i] = 3.4e38f; mini[i] = 0; }

    for (int nb = 0; nb < NT; ++nb) {
#if HAVE_TDM
        if (tid < 32) {                      // wave 0: wait for tile nb to land
            if (nb < NT - 1) __builtin_amdgcn_s_wait_tensorcnt((short)1);
            else             __builtin_amdgcn_s_wait_tensorcnt((short)0);
        }
#else
        {   // fallback: cooperative staging of tile nb
            const float4* src = (const float4*)(cb + (size_t)nb * 16 * CDIM);
            float4* dst = (float4*)&s_cb[nb & 1][0];
            for (int i = tid; i < 16 * CDIM / 4; i += 256) dst[i] = src[i];
        }
#endif
        __syncthreads();

        const float* bBase = &s_cb[nb & 1][bOff];
        v8f acc0 = {};   // two independent WMMA chains for ILP
        v8f acc1 = {};
#pragma unroll 4
        for (int c = 0; c < CDIM; c += 8) {
            v2f a0 = *(const v2f*)(aBase + c);
            v2f b0 = *(const v2f*)(bBase + c);
            acc0 = __builtin_amdgcn_wmma_f32_16x16x4_f32(
                false, a0, false, b0, (short)0, acc0, false, false);
            v2f a1 = *(const v2f*)(aBase + c + 4);
            v2f b1 = *(const v2f*)(bBase + c + 4);
            acc1 = __builtin_amdgcn_wmma_f32_16x16x4_f32(
                false, a1, false, b1, (short)0, acc1, false, false);
        }

        const int   codeIdx = nb * 16 + mn;
        const float e2      = en2[codeIdx];
#pragma unroll
        for (int i = 0; i < 8; ++i) {
            // D layout: VGPR i holds row M=i (lanes 0-15) / M=i+8 (lanes 16-31), N=mn
            float d = e2 - 2.0f * (acc0[i] + acc1[i]);
            if (d < minv[i]) { minv[i] = d; mini[i] = codeIdx; }  // strict < keeps first idx
        }
        __syncthreads();  // all waves done reading buf[nb&1]

#if HAVE_TDM
        if (nb + 2 < NT && tid < 32) {       // refill the buffer just freed
            tdm_load_tile(cb + (size_t)(nb + 2) * 16 * CDIM,
                          (unsigned)(size_t)(void*)&s_cb[nb & 1][0]);
        }
#endif
    }

    // reduce (min,argmin) across the 16 lanes holding the same token row
#pragma unroll
    for (int i = 0; i < 8; ++i) {
#pragma unroll
        for (int s = 1; s < 16; s <<= 1) {
            float ov = __shfl_xor(minv[i], s, 32);
            int   oi = __shfl_xor(mini[i], s, 32);
            if (ov < minv[i] || (ov == minv[i] && oi < mini[i])) {
                minv[i] = ov; mini[i] = oi;
            }
        }
        if (mn == 0) {
            int tok = wg * 128 + (wave << 4) + (hl << 3) + i;
            idx_out[tok]   = mini[i];
            idx_out_f[tok] = (float)mini[i];
        }
    }
}

// ---------- gather z_q, loss partial, scatter EMA sums ----------
__global__ void k_gather(const float* __restrict__ z,
                         const float* __restrict__ cb,
                         const int* __restrict__ idx,
                         float* __restrict__ out_zq,
                         float* __restrict__ out_loss,
                         float* __restrict__ out_cs,
                         float* __restrict__ out_ea) {
    __shared__ float red[256];
    const int t   = blockIdx.x;
    const int tid = threadIdx.x;
    const int k   = idx[t];
    const float* zr = z  + (size_t)t * CDIM;
    const float* cr = cb + (size_t)k * CDIM;
    float* er = out_ea + (size_t)k * CDIM;
    float* oq = out_zq + (size_t)t * CDIM;

    float local = 0.f;
    for (int c = tid; c < CDIM; c += 256) {
        float zv = zr[c], qv = cr[c];
        oq[c] = qv;                               // straight-through fwd value == z_q
        float d = zv - qv;
        local += d * d;
        atomicAdd(er + c, zv * ONE_MINUS_DECAY);  // embed_sum * (1-decay), scattered
    }
    red[tid] = local;
    __syncthreads();
    for (int s = 128; s > 0; s >>= 1) {
        if (tid < s) red[tid] += red[tid + s];
        __syncthreads();
    }
    if (tid == 0) {
        atomicAdd(out_loss, red[0] * (BETAF / ((float)T_TOK * (float)CDIM)));
        atomicAdd(out_cs + k, ONE_MINUS_DECAY);
    }
}

// ---------- n = max(sum(new_cluster_size), 1) ----------
__global__ void k_nsum(const float* __restrict__ out_cs, float* __restrict__ n_out) {
    __shared__ float red[256];
    const int tid = threadIdx.x;
    float s = 0.f;
    for (int i = tid; i < KCODES; i += 256) s += out_cs[i];
    red[tid] = s;
    __syncthreads();
    for (int st = 128; st > 0; st >>= 1) {
        if (tid < st) red[tid] += red[tid + st];
        __syncthreads();
    }
    if (tid == 0) n_out[0] = fmaxf(red[0], 1.0f);
}

// ---------- new_codebook = new_embed_avg / smoothed ----------
__global__ void k_final(const float* __restrict__ out_ea,
                        const float* __restrict__ out_cs,
                        const float* __restrict__ n_ptr,
                        float* __restrict__ out_cb) {
    int i = blockIdx.x * blockDim.x + threadIdx.x;
    if (i >= KCODES * CDIM) return;
    int k = i >> 9;  // / CDIM
    float n = *n_ptr;
    float ncs = out_cs[k];
    float smoothed = (ncs + EPSF) / (n + (float)KCODES * EPSF) * n;
    out_cb[i] = out_ea[i] / smoothed;
}

extern "C" void kernel_launch(void* const* d_in, const int* in_sizes, int n_in,
                              void* d_out, int out_size, void* d_ws, size_t ws_size,
                              hipStream_t stream) {
    const float* z  = (const float*)d_in[0];   // [8,1024,512]
    const float* cb = (const float*)d_in[1];   // [K,C]
    const float* cs = (const float*)d_in[2];   // [K]
    const float* ea = (const float*)d_in[3];   // [K,C]

    float* out      = (float*)d_out;
    float* out_zq   = out;                                   // [T,C]
    float* out_idx  = out + (size_t)T_TOK * CDIM;            // [T] (as float)
    float* out_loss = out_idx + T_TOK;                       // [1]
    float* out_cb   = out_loss + 1;                          // [K,C]
    float* out_cs   = out_cb + (size_t)KCODES * CDIM;        // [K]
    float* out_ea   = out_cs + KCODES;                       // [K,C]

    float* en2   = (float*)d_ws;                 // [K]
    int*   idx   = (int*)(en2 + KCODES);         // [T]
    float* nscal = (float*)(idx + T_TOK);        // [1]

    k_en2<<<KCODES, 32, 0, stream>>>(cb, en2);
    k_init<<<(KCODES * CDIM + 255) / 256, 256, 0, stream>>>(ea, cs, out_ea, out_cs, out_loss);
    k_argmin<<<T_TOK / 128, 256, 0, stream>>>(z, cb, en2, idx, out_idx);
    k_gather<<<T_TOK, 256, 0, stream>>>(z, cb, idx, out_zq, out_loss, out_cs, out_ea);
    k_nsum<<<1, 256, 0, stream>>>(out_cs, nscal);
    k_final<<<(KCODES * CDIM + 255) / 256, 256, 0, stream>>>(out_ea, out_cs, nscal, out_cb);
}